// loTeNet_60988535604001
// MI455X (gfx1250) — compile-verified
//
#include <hip/hip_runtime.h>
#include <math.h>

// ---------------------------------------------------------------------------
// LoTeNet MPO chain on MI455X (gfx1250).
// Bandwidth-bound (~423 MB of single-use weights, AI ~0.5 FLOP/B), so we
// stream Am with coalesced non-temporal b64 loads and use the exact-f32
// matrix engine V_WMMA_F32_16X16X4_F32 to build P = M1*...*M7 per pixel.
// One wave32 == one pixel == one workgroup (<=32 single-wave WGs per WGP).
// Embedding uses the native v_sin/v_cos units (9 scalars/pixel) instead of
// the libm Payne-Hanek blob that dominated the previous build's SALU count.
// ---------------------------------------------------------------------------

typedef float v2f __attribute__((ext_vector_type(2)));
typedef float v8f __attribute__((ext_vector_type(8)));

#define XS 33  // padded LDS row stride (coprime with 64 banks)

__device__ __forceinline__ v2f ldnt_v2(const float* p) {
    // streaming weights: used exactly once -> non-temporal hint
    return __builtin_nontemporal_load((const v2f*)p);
}

// One block = 32 threads = one wave = one pixel n.
//   img  : input image of side Sin (Sin = 3*S), feature f = img[3h+f/3, 3w+f%3]
//   Af   : (N, 32, 2, OO)   Am : (N, 7, 32, 32, 2)   Al : (N, 32, 2)
//   out  : OO==1 -> out[n] scalar per pixel; OO==10 -> out[0..9]
__global__ __launch_bounds__(32) void mpo_layer(
    const float* __restrict__ img, int Sin,
    const float* __restrict__ Af, const float* __restrict__ Am,
    const float* __restrict__ Al,
    float* __restrict__ out, int S, int OO)
{
    __shared__ float X[32 * XS];   // running product, row-major [d][e]
    __shared__ float Ash[32];      // boundary vector a (OO==1 path)
    __shared__ float Ush[32];      // boundary vector u
    __shared__ float Red[32];      // reduction buffer

    const int n    = blockIdx.x;
    const int lane = threadIdx.x;
    const int h = n / S, w = n % S;

    // --- squeeze + trig embedding (uniform across lanes, 9 scalars) ----------
    // native v_sin_f32 / v_cos_f32: tiny code, ample accuracy for cos/sin(pi/2*x)
    float c[9], s[9];
#pragma unroll
    for (int f = 0; f < 9; ++f) {
        float val = img[(size_t)(3 * h + f / 3) * Sin + (3 * w + f % 3)];
        float ang = 1.57079632679489662f * val;   // pi/2 * x
        s[f] = __sinf(ang);
        c[f] = __cosf(ang);
    }

    // --- boundary vectors ----------------------------------------------------
    {
        const float* al = Al + ((size_t)n * 32 + lane) * 2;
        Ush[lane] = al[0] * c[8] + al[1] * s[8];
        // Ash uses the OO==1 layout; for the head (OO==10) it is unused but the
        // read stays in-bounds (AfL has 640 floats). Unconditional => uniform EXEC.
        const float* af = Af + ((size_t)n * 32 + lane) * 2;
        Ash[lane] = af[0] * c[0] + af[1] * s[0];
    }

    // --- X = M1 (Am site 0, feature 1), row-major into LDS -------------------
    const size_t amStride = 7ull * 32 * 32 * 2;
    {
        const float* m0 = Am + (size_t)n * amStride;
#pragma unroll 4
        for (int d = 0; d < 32; ++d) {
            v2f p = ldnt_v2(m0 + ((size_t)d * 32 + lane) * 2);  // coalesced 256B
            X[d * XS + lane] = p.x * c[1] + p.y * s[1];
        }
    }
    __syncthreads();

    // --- chain X <- X * M_{j+1}  (Am site j, feature j+1), j = 1..6 ----------
    // f32 WMMA 16x16x4 operand layouts (CDNA5 ISA):
    //   A (16x4): lane L, vgpr v -> [m = L%16,            k = v + 2*(L/16)]
    //   B (4x16): lane L, vgpr v -> [k = v + 2*(L/16),    nn = L%16]
    //   C (16x16): lane L, vgpr r -> [row = r + 8*(L/16), col = L%16]
    const int mA    = lane & 15;
    const int half2 = (lane >> 4) * 2;   // 0 or 2
    const int e0    = lane & 15;

#pragma unroll
    for (int j = 1; j < 7; ++j) {        // fully unrolled: constant c/s indices,
        const float cs = c[j + 1];       // and next-site global loads can be
        const float ss = s[j + 1];       // hoisted above this site's writeback
        const float* mb = Am + (size_t)n * amStride + (size_t)j * 2048;

        v8f acc00 = {}, acc01 = {}, acc10 = {}, acc11 = {};
#pragma unroll
        for (int k8 = 0; k8 < 8; ++k8) {
            const int kA = 4 * k8 + half2;
            // A chunks of X from LDS (padded stride -> conflict-free)
            v2f A0 = { X[mA * XS + kA],        X[mA * XS + kA + 1] };
            v2f A1 = { X[(mA + 16) * XS + kA], X[(mA + 16) * XS + kA + 1] };
            // B chunks of M straight from HBM: lanes 0..15 consecutive in e
            // -> coalesced b64 (cos,sin pair); combine with per-site scalars.
            v2f q00 = ldnt_v2(mb + ((size_t)(kA + 0) * 32 + e0) * 2);
            v2f q01 = ldnt_v2(mb + ((size_t)(kA + 1) * 32 + e0) * 2);
            v2f q10 = ldnt_v2(mb + ((size_t)(kA + 0) * 32 + e0 + 16) * 2);
            v2f q11 = ldnt_v2(mb + ((size_t)(kA + 1) * 32 + e0 + 16) * 2);
            v2f B0 = { q00.x * cs + q00.y * ss, q01.x * cs + q01.y * ss };
            v2f B1 = { q10.x * cs + q10.y * ss, q11.x * cs + q11.y * ss };

            acc00 = __builtin_amdgcn_wmma_f32_16x16x4_f32(false, A0, false, B0,
                                                          (short)0, acc00, false, false);
            acc01 = __builtin_amdgcn_wmma_f32_16x16x4_f32(false, A0, false, B1,
                                                          (short)0, acc01, false, false);
            acc10 = __builtin_amdgcn_wmma_f32_16x16x4_f32(false, A1, false, B0,
                                                          (short)0, acc10, false, false);
            acc11 = __builtin_amdgcn_wmma_f32_16x16x4_f32(false, A1, false, B1,
                                                          (short)0, acc11, false, false);
        }
        __syncthreads();
        // write D (C layout) back to row-major LDS
        const int rbase = 8 * (lane >> 4);
#pragma unroll
        for (int r = 0; r < 8; ++r) {
            X[(rbase + r) * XS + e0]           = acc00[r];
            X[(rbase + r) * XS + e0 + 16]      = acc01[r];
            X[(rbase + r + 16) * XS + e0]      = acc10[r];
            X[(rbase + r + 16) * XS + e0 + 16] = acc11[r];
        }
        __syncthreads();
    }

    // --- final contraction out = a * P * u -----------------------------------
    if (OO == 1) {
        float t = 0.f;
#pragma unroll 4
        for (int d = 0; d < 32; ++d) t += Ash[d] * X[d * XS + lane];  // row bcast
        Red[lane] = t * Ush[lane];
        __syncthreads();
        if (lane == 0) {
            float acc = 0.f;
            for (int e = 0; e < 32; ++e) acc += Red[e];
            out[n] = acc;
        }
    } else {
        // head: v[d] = sum_e P[d,e]*u[e]; out[o] = sum_d aL[o,d]*v[d]
        float vd = 0.f;
#pragma unroll 4
        for (int e = 0; e < 32; ++e) vd += X[lane * XS + e] * Ush[e];
        Red[lane] = vd;
        __syncthreads();
        if (lane < OO) {
            float acc = 0.f;
            for (int d = 0; d < 32; ++d) {
                // AfL: (1, 32, 2, OO) -> [d,p,o] = d*2*OO + p*OO + o
                float a = Af[(size_t)d * 2 * OO + lane] * c[0]
                        + Af[(size_t)d * 2 * OO + OO + lane] * s[0];
                acc += a * Red[d];
            }
            out[lane] = acc;
        }
    }
}

extern "C" void kernel_launch(void* const* d_in, const int* in_sizes, int n_in,
                              void* d_out, int out_size, void* d_ws, size_t ws_size,
                              hipStream_t stream) {
    (void)in_sizes; (void)n_in; (void)out_size; (void)ws_size;
    const float* image = (const float*)d_in[0];

    // inter-layer scalar images in workspace
    float* v0 = (float*)d_ws;          // 6561
    float* v1 = v0 + 6561;             // 729
    float* v2 = v1 + 729;              // 81
    float* v3 = v2 + 81;               // 9

    // layer 0: 243 -> 81
    mpo_layer<<<6561, 32, 0, stream>>>(image, 243,
        (const float*)d_in[1], (const float*)d_in[2], (const float*)d_in[3],
        v0, 81, 1);
    // layer 1: 81 -> 27
    mpo_layer<<<729, 32, 0, stream>>>(v0, 81,
        (const float*)d_in[4], (const float*)d_in[5], (const float*)d_in[6],
        v1, 27, 1);
    // layer 2: 27 -> 9
    mpo_layer<<<81, 32, 0, stream>>>(v1, 27,
        (const float*)d_in[7], (const float*)d_in[8], (const float*)d_in[9],
        v2, 9, 1);
    // layer 3: 9 -> 3
    mpo_layer<<<9, 32, 0, stream>>>(v2, 9,
        (const float*)d_in[10], (const float*)d_in[11], (const float*)d_in[12],
        v3, 3, 1);
    // head: 3 -> 1, OUT=10
    mpo_layer<<<1, 32, 0, stream>>>(v3, 3,
        (const float*)d_in[13], (const float*)d_in[14], (const float*)d_in[15],
        (float*)d_out, 1, 10);
}